// Connector_22849226015383
// MI455X (gfx1250) — compile-verified
//
#include <hip/hip_runtime.h>
#include <hip/hip_bf16.h>

// MI455X / gfx1250. Two-pass bandwidth-bound implementation:
//   K1: per-16-token tile, WMMA f32 16x16x4 GEMM vs phi (+sumsq), Sinkhorn -> ws
//   K2: streaming out = M @ residual + H (x) output  (NT stores)
typedef float v2f __attribute__((ext_vector_type(2)));
typedef float v4f __attribute__((ext_vector_type(4)));
typedef float v8f __attribute__((ext_vector_type(8)));

static constexpr int kB = 4;
static constexpr int kS = 2048;
static constexpr int kN = 4;
static constexpr int kC = 2048;
static constexpr int kNC = kN * kC;          // 8192
static constexpr int kTokens = kB * kS;      // 8192
static constexpr int kSinkIters = 20;
static constexpr float kEps = 1e-5f;
static constexpr int kWaves = 8;             // 256 threads, wave32
static constexpr int kKPerWave = kNC / kWaves; // 1024

// ---------------------------------------------------------------------------
// Kernel 1: per-token stats via WMMA, then Sinkhorn + sigmoid.
// Grid: kTokens/16 blocks of 256 threads. Each block owns 16 tokens.
// ---------------------------------------------------------------------------
__global__ __launch_bounds__(256)
void connector_stats_kernel(const float* __restrict__ residual,   // [tok][8192]
                            const float* __restrict__ rms_scale,  // [8192]
                            const float* __restrict__ phi_post,   // [8192][4]
                            const float* __restrict__ phi_res,    // [8192][16]
                            const float* __restrict__ b_post,     // [4]
                            const float* __restrict__ b_res,      // [4][4]
                            const float* __restrict__ alpha_post_p,
                            const float* __restrict__ alpha_res_p,
                            float* __restrict__ mh)               // [tok][20]
{
    __shared__ float ldsR[kWaves][16][16];   // phi_res partial dots (D tiles)
    __shared__ float ldsP[kWaves][16][4];    // phi_post partial dots
    __shared__ float ldsSS[kWaves][32];      // sum-of-squares partials

    const int tid  = threadIdx.x;
    const int wave = tid >> 5;
    const int lane = tid & 31;
    const int half = lane >> 4;              // selects K pair within wmma tile
    const int col  = lane & 15;              // A: token row / B,D: column
    const int tokBase = blockIdx.x * 16;

    const float* resRow = residual + (size_t)(tokBase + col) * kNC;

    // phi_post mask: cols 0..3 live, rest zero. Address is clamped so every
    // lane issues a plain load (no EXEC-masked branches in the hot loop).
    const float pmask = (col < 4) ? 1.0f : 0.0f;
    const int   pcol  = col & 3;

    v8f accR = {};    // 16x16 D: rows = tokens, cols = phi_res columns
    v8f accP = {};    // cols 0..3 = phi_post columns, rest junk
    float ssq = 0.f;

    const int kbase = wave * kKPerWave;
    #pragma unroll 4
    for (int ki = 0; ki < kKPerWave; ki += 4) {
        const int ka = kbase + ki + 2 * half;          // this lane's K pair
        // A fragment: raw residual, then apply rms_scale (rsqrt factored out).
        v2f a  = *(const v2f*)(resRow + ka);
        v2f sc = *(const v2f*)(rms_scale + ka);
        ssq += a.x * a.x + a.y * a.y;
        v2f as; as.x = a.x * sc.x; as.y = a.y * sc.y;
        // B fragment for phi_res: B[k][n], n = col.
        v2f br;
        br.x = phi_res[(size_t)ka * 16 + col];
        br.y = phi_res[(size_t)(ka + 1) * 16 + col];
        // B fragment for phi_post: always-in-bounds load, masked by VALU mul.
        v2f bp;
        bp.x = pmask * phi_post[(size_t)ka * 4 + pcol];
        bp.y = pmask * phi_post[(size_t)(ka + 1) * 4 + pcol];
        // D += A * B  (16x16x4 f32)
        accR = __builtin_amdgcn_wmma_f32_16x16x4_f32(
            false, as, false, br, (short)0, accR, false, false);
        accP = __builtin_amdgcn_wmma_f32_16x16x4_f32(
            false, as, false, bp, (short)0, accP, false, false);
    }

    // D layout: element (vgpr v, lane) = D[M = v + 8*half][N = col]
    #pragma unroll
    for (int v = 0; v < 8; ++v) {
        ldsR[wave][v + 8 * half][col] = accR[v];
        if (col < 4) ldsP[wave][v + 8 * half][col] = accP[v];
    }
    ldsSS[wave][lane] = ssq;
    __syncthreads();

    // 16 lanes finalize one token each: reduce across waves, Sinkhorn, sigmoid.
    if (tid < 16) {
        const int t = tid;
        float Gres[16], Gpost[4], ss = 0.f;
        #pragma unroll
        for (int c = 0; c < 16; ++c) Gres[c] = 0.f;
        #pragma unroll
        for (int c = 0; c < 4; ++c) Gpost[c] = 0.f;
        for (int w = 0; w < kWaves; ++w) {
            ss += ldsSS[w][t] + ldsSS[w][t + 16];
            #pragma unroll
            for (int c = 0; c < 16; ++c) Gres[c] += ldsR[w][t][c];
            #pragma unroll
            for (int c = 0; c < 4; ++c) Gpost[c] += ldsP[w][t][c];
        }
        const float rnorm = rsqrtf(ss * (1.0f / kNC) + kEps);
        const float aR = alpha_res_p[0];
        const float aP = alpha_post_p[0];

        float T[4][4];
        #pragma unroll
        for (int i = 0; i < 4; ++i)
            #pragma unroll
            for (int j = 0; j < 4; ++j)
                T[i][j] = aR * rnorm * Gres[i * 4 + j] + b_res[i * 4 + j];

        for (int it = 0; it < kSinkIters; ++it) {
            #pragma unroll
            for (int i = 0; i < 4; ++i) {     // row logsumexp
                float m = fmaxf(fmaxf(T[i][0], T[i][1]), fmaxf(T[i][2], T[i][3]));
                float s = __expf(T[i][0] - m) + __expf(T[i][1] - m) +
                          __expf(T[i][2] - m) + __expf(T[i][3] - m);
                float l = m + __logf(s);
                #pragma unroll
                for (int j = 0; j < 4; ++j) T[i][j] -= l;
            }
            #pragma unroll
            for (int j = 0; j < 4; ++j) {     // col logsumexp
                float m = fmaxf(fmaxf(T[0][j], T[1][j]), fmaxf(T[2][j], T[3][j]));
                float s = __expf(T[0][j] - m) + __expf(T[1][j] - m) +
                          __expf(T[2][j] - m) + __expf(T[3][j] - m);
                float l = m + __logf(s);
                #pragma unroll
                for (int i = 0; i < 4; ++i) T[i][j] -= l;
            }
        }

        float* o = mh + (size_t)(tokBase + t) * 20;
        #pragma unroll
        for (int i = 0; i < 4; ++i)
            #pragma unroll
            for (int j = 0; j < 4; ++j)
                o[i * 4 + j] = __expf(T[i][j]);
        #pragma unroll
        for (int n = 0; n < 4; ++n) {
            float x = aP * rnorm * Gpost[n] + b_post[n];
            o[16 + n] = 2.0f / (1.0f + __expf(-x));
        }
    }
}

// ---------------------------------------------------------------------------
// Kernel 2: out[t][n][c] = sum_k M[t][n][k]*res[t][k][c] + H[t][n]*output[t][c]
// Grid: kTokens*2 blocks of 256 threads; each thread handles a v4f chunk.
// Result is write-once/never-read: use non-temporal stores to keep L2 free
// for the residual stream.
// ---------------------------------------------------------------------------
__global__ __launch_bounds__(256)
void connector_apply_kernel(const float* __restrict__ residual, // [tok][4][2048]
                            const float* __restrict__ outp,     // [tok][2048]
                            const float* __restrict__ mh,       // [tok][20]
                            float* __restrict__ out)            // [tok][4][2048]
{
    const int tid   = threadIdx.x;
    const int part  = blockIdx.x & 1;
    const int token = blockIdx.x >> 1;
    const int c = part * 1024 + tid * 4;

    __shared__ float sMH[20];
    if (tid < 20) sMH[tid] = mh[(size_t)token * 20 + tid];
    __syncthreads();

    const float* resRow = residual + (size_t)token * kNC;
    const v4f r0 = *(const v4f*)(resRow + 0 * kC + c);
    const v4f r1 = *(const v4f*)(resRow + 1 * kC + c);
    const v4f r2 = *(const v4f*)(resRow + 2 * kC + c);
    const v4f r3 = *(const v4f*)(resRow + 3 * kC + c);
    const v4f ov = *(const v4f*)(outp + (size_t)token * kC + c);

    float* outRow = out + (size_t)token * kNC;
    #pragma unroll
    for (int n = 0; n < 4; ++n) {
        const float m0 = sMH[n * 4 + 0], m1 = sMH[n * 4 + 1];
        const float m2 = sMH[n * 4 + 2], m3 = sMH[n * 4 + 3];
        const float h  = sMH[16 + n];
        v4f r;
        r.x = m0 * r0.x + m1 * r1.x + m2 * r2.x + m3 * r3.x + h * ov.x;
        r.y = m0 * r0.y + m1 * r1.y + m2 * r2.y + m3 * r3.y + h * ov.y;
        r.z = m0 * r0.z + m1 * r1.z + m2 * r2.z + m3 * r3.z + h * ov.z;
        r.w = m0 * r0.w + m1 * r1.w + m2 * r2.w + m3 * r3.w + h * ov.w;
        __builtin_nontemporal_store(r, (v4f*)(outRow + n * kC + c));
    }
}

// ---------------------------------------------------------------------------
extern "C" void kernel_launch(void* const* d_in, const int* in_sizes, int n_in,
                              void* d_out, int out_size, void* d_ws, size_t ws_size,
                              hipStream_t stream) {
    const float* residual   = (const float*)d_in[0];
    const float* output     = (const float*)d_in[1];
    const float* rms_scale  = (const float*)d_in[2];
    const float* phi_post   = (const float*)d_in[3];
    const float* phi_res    = (const float*)d_in[4];
    const float* b_post     = (const float*)d_in[5];
    const float* b_res      = (const float*)d_in[6];
    const float* alpha_post = (const float*)d_in[7];
    const float* alpha_res  = (const float*)d_in[8];
    float* mh = (float*)d_ws;                 // kTokens * 20 floats = 640 KB

    connector_stats_kernel<<<kTokens / 16, 256, 0, stream>>>(
        residual, rms_scale, phi_post, phi_res, b_post, b_res,
        alpha_post, alpha_res, mh);
    connector_apply_kernel<<<kTokens * 2, 256, 0, stream>>>(
        residual, output, mh, (float*)d_out);
}